// GNNDiscLayer_5944234737797
// MI455X (gfx1250) — compile-verified
//
#include <hip/hip_runtime.h>
#include <hip/hip_bf16.h>

typedef __attribute__((ext_vector_type(2))) float v2f;
typedef __attribute__((ext_vector_type(4))) float v4f;
typedef __attribute__((ext_vector_type(8))) float v8f;

#define D_FEAT 64

// h = in @ W^T + b. in: N x 64 (row-major), W: 64 x 64 (row-major), h: N x 64.
// One wave32 computes a 16-node x 64-feature block using V_WMMA_F32_16X16X4_F32.
// A (16x4, MxK):  lane = m + 16*(k>>1), vgpr = k&1   -> per-lane float2 of in-row
// B (4x16, KxN):  lane = n + 16*(k>>1), vgpr = k&1   -> per-lane float2 of W-row
// C (16x16 f32):  vgpr v holds (M = v + 8*(lane>>4), N = lane&15)
__global__ __launch_bounds__(256) void gnn_gemm_kernel(
    const float* __restrict__ inA, const float* __restrict__ inB,
    const int* __restrict__ depth, int selectByDepth,
    const float* __restrict__ W, const float* __restrict__ b,
    float* __restrict__ h, int N)
{
  const float* in = (selectByDepth && depth[0] != 1) ? inB : inA;
  const int lane = threadIdx.x & 31;
  const int wave = blockIdx.x * (blockDim.x >> 5) + (threadIdx.x >> 5);
  const int nodeBase = wave * 16;
  if (nodeBase >= N) return;           // wave-uniform: EXEC stays all-ones for WMMA
  const int lo = lane & 15;            // m (for A) / n (for B, C)
  const int hi = lane >> 4;            // which half of the K quad
  int rowA = nodeBase + lo;
  if (rowA >= N) rowA = N - 1;         // clamp (no effect when N % 16 == 0)
  const float* arow = in + (size_t)rowA * D_FEAT + 2 * hi;
  const bool fullTile = (nodeBase + 16 <= N);   // wave-uniform

  for (int jt = 0; jt < 4; ++jt) {
    const int jBase = jt * 16;
    const float bj = b[jBase + lo];
    v8f c;
#pragma unroll
    for (int v = 0; v < 8; ++v) c[v] = bj;   // C init = bias broadcast along N
    const float* brow = W + (size_t)(jBase + lo) * D_FEAT + 2 * hi;
#pragma unroll
    for (int kt = 0; kt < 16; ++kt) {        // K = 64 in steps of 4
      v2f a  = *(const v2f*)(arow + kt * 4);
      v2f bb = *(const v2f*)(brow + kt * 4);
      c = __builtin_amdgcn_wmma_f32_16x16x4_f32(
          /*neg_a=*/false, a, /*neg_b=*/false, bb,
          /*c_mod=*/(short)0, c, /*reuse_a=*/false, /*reuse_b=*/false);
    }
    float* hbase = h + (size_t)(nodeBase + 8 * hi) * D_FEAT + jBase + lo;
    if (fullTile) {
      // common path: unguarded stores, no per-element EXEC juggling
#pragma unroll
      for (int v = 0; v < 8; ++v) hbase[(size_t)v * D_FEAT] = c[v];
    } else {
#pragma unroll
      for (int v = 0; v < 8; ++v) {
        int row = nodeBase + v + 8 * hi;
        if (row < N) h[(size_t)row * D_FEAT + jBase + lo] = c[v];
      }
    }
  }
}

// 16 threads per edge, float4 per thread. Gathers hit L2 (node arrays ~13MB each);
// the big streaming `output` write is non-temporal to keep L2 for node data.
__global__ __launch_bounds__(256) void gnn_edge_kernel(
    const float* __restrict__ root, const float* __restrict__ m,
    const float* __restrict__ x, const float* __restrict__ hmsg,
    const int* __restrict__ src, const int* __restrict__ dst,
    float* __restrict__ out, float* __restrict__ sum_msg,
    float* __restrict__ sum_root, float* __restrict__ deg, int E)
{
  long long t = (long long)blockIdx.x * blockDim.x + threadIdx.x;
  int e = (int)(t >> 4);
  if (e >= E) return;
  int f = ((int)t & 15) * 4;
  int s = src[e], d = dst[e];

  v4f r  = *(const v4f*)(root + (size_t)s * D_FEAT + f);
  v4f mm = *(const v4f*)(m    + (size_t)s * D_FEAT + f);
  v4f xx = *(const v4f*)(x    + (size_t)d * D_FEAT + f);

  float* o = out + (size_t)e * (3 * D_FEAT);
  __builtin_nontemporal_store(r,  (v4f*)(o + f));
  __builtin_nontemporal_store(mm, (v4f*)(o + D_FEAT + f));
  __builtin_nontemporal_store(xx, (v4f*)(o + 2 * D_FEAT + f));

  v4f hh = *(const v4f*)(hmsg + (size_t)s * D_FEAT + f);
  float* sm = sum_msg  + (size_t)d * D_FEAT + f;
  float* sr = sum_root + (size_t)d * D_FEAT + f;
#pragma unroll
  for (int i = 0; i < 4; ++i) {
    atomicAdd(sm + i, hh[i]);
    atomicAdd(sr + i, r[i]);
  }
  if ((t & 15) == 0) atomicAdd(deg + d, 1.0f);
}

// Per-node finalize: m_new = deg>0 ? relu(h_self + sum_msg/deg) : m
//                    root_new = deg>0 ? sum_root/deg : root
__global__ __launch_bounds__(256) void gnn_finalize_kernel(
    const float* __restrict__ hself, const float* __restrict__ sum_msg,
    const float* __restrict__ sum_root, const float* __restrict__ m,
    const float* __restrict__ root, const float* __restrict__ deg,
    float* __restrict__ m_new, float* __restrict__ root_new, int N)
{
  int t = blockIdx.x * blockDim.x + threadIdx.x;
  int n = t >> 4;
  if (n >= N) return;
  int f = (t & 15) * 4;
  float dg = deg[n];
  v4f mo, ro;
  if (dg > 0.0f) {
    float inv = 1.0f / dg;
    v4f hs = *(const v4f*)(hself   + (size_t)n * D_FEAT + f);
    v4f sm = *(const v4f*)(sum_msg + (size_t)n * D_FEAT + f);
    v4f sr = *(const v4f*)(sum_root+ (size_t)n * D_FEAT + f);
#pragma unroll
    for (int i = 0; i < 4; ++i) {
      float v = hs[i] + sm[i] * inv;
      mo[i] = v > 0.0f ? v : 0.0f;
      ro[i] = sr[i] * inv;
    }
  } else {
    mo = *(const v4f*)(m    + (size_t)n * D_FEAT + f);
    ro = *(const v4f*)(root + (size_t)n * D_FEAT + f);
  }
  *(v4f*)(m_new    + (size_t)n * D_FEAT + f) = mo;
  *(v4f*)(root_new + (size_t)n * D_FEAT + f) = ro;
}

extern "C" void kernel_launch(void* const* d_in, const int* in_sizes, int n_in,
                              void* d_out, int out_size, void* d_ws, size_t ws_size,
                              hipStream_t stream) {
  const float* x    = (const float*)d_in[0];
  const float* root = (const float*)d_in[1];
  const float* m    = (const float*)d_in[2];
  const float* W    = (const float*)d_in[3];
  const float* b    = (const float*)d_in[4];
  const int*   src  = (const int*)d_in[5];
  const int*   dst  = (const int*)d_in[6];
  const int*   depth= (const int*)d_in[7];

  const int N = in_sizes[0] / D_FEAT;   // 50000
  const int E = in_sizes[5];            // 800000

  // Workspace layout (floats): h_self | h_msg | sum_msg | sum_root | deg
  float* ws       = (float*)d_ws;
  float* h_self   = ws;
  float* h_msg    = h_self   + (size_t)N * D_FEAT;
  float* sum_msg  = h_msg    + (size_t)N * D_FEAT;
  float* sum_root = sum_msg  + (size_t)N * D_FEAT;
  float* deg      = sum_root + (size_t)N * D_FEAT;

  // Output layout (floats): output[E*192] | m_new[N*64] | root_new[N*64]
  float* out      = (float*)d_out;
  float* m_new    = out   + (size_t)E * (3 * D_FEAT);
  float* root_new = m_new + (size_t)N * D_FEAT;

  // Zero accumulators (sum_msg, sum_root, deg are contiguous)
  hipMemsetAsync(sum_msg, 0, ((size_t)2 * N * D_FEAT + N) * sizeof(float), stream);

  // Two tiny WMMA GEMMs: h_self = x@W^T+b; h_msg = (depth==1 ? x : m)@W^T+b
  const int waves  = (N + 15) / 16;
  const int gBlocks = (waves + 7) / 8;          // 8 waves (256 threads) per block
  gnn_gemm_kernel<<<gBlocks, 256, 0, stream>>>(x, x, depth, 0, W, b, h_self, N);
  gnn_gemm_kernel<<<gBlocks, 256, 0, stream>>>(x, m, depth, 1, W, b, h_msg,  N);

  // Edge gather/concat + scatter-add reductions
  const long long eThreads = (long long)E * 16;
  const int eBlocks = (int)((eThreads + 255) / 256);
  gnn_edge_kernel<<<eBlocks, 256, 0, stream>>>(root, m, x, h_msg, src, dst,
                                               out, sum_msg, sum_root, deg, E);

  // Node finalize
  const int fBlocks = (N * 16 + 255) / 256;
  gnn_finalize_kernel<<<fBlocks, 256, 0, stream>>>(h_self, sum_msg, sum_root,
                                                   m, root, deg, m_new, root_new, N);
}